// Linear_regression_67723044323563
// MI455X (gfx1250) — compile-verified
//
#include <hip/hip_runtime.h>

// ---------------------------------------------------------------------------
// out0 = dot(w[0:W], x) + dot(w[W:2W], x*x) + w[2W]
// out1 = x[W/2] - out0
// Pure streaming reduction: 192 MB traffic -> ~8.3 us floor at 23.3 TB/s.
// Pass 1: 4096 blocks stream x / w_lin / w_quad with b128 loads, fp32 FMAs,
//         wave32 shfl reduction + LDS, one partial per block into d_ws.
// Pass 2: single wave32 sums the 4096 partials, then collapses the 32 lane
//         sums with one V_WMMA_F32_16X16X4_F32 (B = ones), exact fp32 math.
// ---------------------------------------------------------------------------

typedef __attribute__((ext_vector_type(2))) float v2f;
typedef __attribute__((ext_vector_type(8))) float v8f;

#define W_ELEMS 16777216
#define NPART   4096
#define TPB     256

__global__ __launch_bounds__(TPB)
void lr_partial_kernel(const float* __restrict__ x,
                       const float* __restrict__ w,
                       float* __restrict__ partials) {
    const float4* __restrict__ x4  = (const float4*)x;
    const float4* __restrict__ wl4 = (const float4*)w;
    const float4* __restrict__ wq4 = (const float4*)(w + W_ELEMS);

    const int n4     = W_ELEMS / 4;          // 4,194,304 float4 per stream
    const int stride = NPART * TPB;          // 1,048,576 lanes

    float acc = 0.0f;
    #pragma unroll 4
    for (int i = blockIdx.x * TPB + threadIdx.x; i < n4; i += stride) {
        float4 xv = x4[i];
        float4 lv = wl4[i];
        float4 qv = wq4[i];
        // linear term
        acc = fmaf(lv.x, xv.x, acc);
        acc = fmaf(lv.y, xv.y, acc);
        acc = fmaf(lv.z, xv.z, acc);
        acc = fmaf(lv.w, xv.w, acc);
        // quadratic term
        acc = fmaf(qv.x, xv.x * xv.x, acc);
        acc = fmaf(qv.y, xv.y * xv.y, acc);
        acc = fmaf(qv.z, xv.z * xv.z, acc);
        acc = fmaf(qv.w, xv.w * xv.w, acc);
    }

    // wave32 reduction
    #pragma unroll
    for (int off = 16; off > 0; off >>= 1)
        acc += __shfl_xor(acc, off, 32);

    __shared__ float red[TPB / 32];
    const int wid  = threadIdx.x >> 5;
    const int lane = threadIdx.x & 31;
    if (lane == 0) red[wid] = acc;
    __syncthreads();

    if (threadIdx.x == 0) {
        float s = red[0];
        #pragma unroll
        for (int i = 1; i < TPB / 32; ++i) s += red[i];
        partials[blockIdx.x] = s;
    }
}

__global__ __launch_bounds__(32)
void lr_finalize_kernel(const float* __restrict__ partials,
                        const float* __restrict__ x,
                        const float* __restrict__ w,
                        float* __restrict__ out) {
    const int lane = threadIdx.x;   // single wave32, EXEC all ones

    // 4096 partials / 32 lanes = 128 sequential fp32 adds per lane (L2-hot)
    float s = 0.0f;
    for (int i = lane; i < NPART; i += 32)
        s += partials[i];

    // Collapse 32 lane sums with one fp32 WMMA:
    //   A (16x4): lane value in VGPR0 -> A[m][0] (lanes 0-15), A[m][2]
    //   (lanes 16-31); VGPR1 = 0.  B (4x16) = all ones.
    //   D[m][n] = sum_k A[m][k] = val[m] + val[m+16]  (independent of n)
    // Summing D's 8 VGPRs on a lane gives the half-total for its lane group;
    // adding the xor-16 partner gives the exact full total.
    v2f a; a.x = s;    a.y = 0.0f;
    v2f b; b.x = 1.0f; b.y = 1.0f;
    v8f c = {};
    v8f d = __builtin_amdgcn_wmma_f32_16x16x4_f32(
        /*neg_a=*/false, a, /*neg_b=*/false, b,
        /*c_mod=*/(short)0, c, /*reuse_a=*/false, /*reuse_b=*/false);

    float t = 0.0f;
    #pragma unroll
    for (int i = 0; i < 8; ++i) t += d[i];
    float total = t + __shfl_xor(t, 16, 32);

    if (lane == 0) {
        float out0 = total + w[2 * W_ELEMS];
        out[0] = out0;
        out[1] = x[W_ELEMS / 2] - out0;
    }
}

extern "C" void kernel_launch(void* const* d_in, const int* in_sizes, int n_in,
                              void* d_out, int out_size, void* d_ws, size_t ws_size,
                              hipStream_t stream) {
    const float* x = (const float*)d_in[0];   // W floats
    const float* w = (const float*)d_in[1];   // 2W+1 floats
    float* out      = (float*)d_out;          // 2 floats
    float* partials = (float*)d_ws;           // NPART floats (16 KB)

    lr_partial_kernel<<<NPART, TPB, 0, stream>>>(x, w, partials);
    lr_finalize_kernel<<<1, 32, 0, stream>>>(partials, x, w, out);
}